// QuadraticTimeVaryingVF_42142219108830
// MI455X (gfx1250) — compile-verified
//
#include <hip/hip_runtime.h>

typedef __attribute__((ext_vector_type(2))) float v2f;
typedef __attribute__((ext_vector_type(8))) float v8f;

// Problem shape (fixed by the reference)
static constexpr int kH    = 64;    // horizon
static constexpr int kD    = 64;    // d_obs
static constexpr int kDin  = 2145;  // 64 + 2080 + 1
static constexpr int kRowsPerWave = 16;
static constexpr int kWaves       = 8;   // 256 threads/block
static constexpr int kKChunks     = 17;  // K = 68 = 64 obs + 1 const + 3 zero-pad
static constexpr int kBpStride    = 65;  // in float2 units (bank-conflict padding)
static constexpr int kXsStride    = 68;  // floats (conflict-free b64 store pattern)

__global__ __launch_bounds__(256) void qvf_wmma_kernel(
    const float* __restrict__ obs,   // [N, H, D]
    const float* __restrict__ W,     // [H, 2145]
    const float* __restrict__ bias,  // [H]
    float* __restrict__ out)         // [N, H]
{
  // B-matrix for this h, K=68 x N=64, stored as (K-pair, i) float2 fragments:
  // Bp[(j>>1)*kBpStride + i] = { Bmat[j0][i], Bmat[j0+1][i] }
  __shared__ v2f   Bp[34 * kBpStride];                  // ~17.7 KB
  __shared__ float Xs[kWaves][kRowsPerWave * kXsStride]; // ~34.8 KB

  const int tid  = threadIdx.x;
  const int h    = blockIdx.y;
  const int wave = tid >> 5;
  const int lane = tid & 31;
  const int lo   = lane & 15;   // N / M sub-index
  const int hi   = lane >> 4;   // selects K pair {0,1} vs {2,3}

  // ---- cooperative unpack of weights[h] into Bmat (upper-tri + linear row) ----
  {
    float* Bf = (float*)Bp;
    const float* Wh = W + (size_t)h * kDin;
    #pragma unroll
    for (int e = tid; e < 68 * 64; e += 256) {   // 17 iters, uniform
      const int j = e >> 6;   // K index 0..67
      const int i = e & 63;   // output-feature index
      float v = 0.f;
      if (j < 64) {
        // packed triu order: idx(i,j) = 64i - i(i-1)/2 + (j-i), i<=j
        if (i <= j) v = Wh[64 + i * 64 - (i * (i - 1)) / 2 + (j - i)];
      } else if (j == 64) {
        v = Wh[i];            // linear term, multiplied by constant x[64]=1
      }                       // j = 65..67: zero padding
      Bf[((j >> 1) * kBpStride + i) * 2 + (j & 1)] = v;
    }
  }

  // ---- load X tile as WMMA A-fragments (16x4 f32 layout) + stage to LDS ----
  const int n_base = blockIdx.x * (kWaves * kRowsPerWave) + wave * kRowsPerWave;
  const int m    = lo;        // matrix row M held by this lane
  const int koff = hi * 2;    // K offset inside each 4-chunk
  const float* xrow = obs + ((size_t)(n_base + m) * kH + h) * kD;

  v2f a[kKChunks];
  #pragma unroll
  for (int kk = 0; kk < 16; ++kk) {
    v2f t = *(const v2f*)(xrow + 4 * kk + koff);
    a[kk] = t;
    *(v2f*)(&Xs[wave][m * kXsStride + 4 * kk + koff]) = t;
  }
  {
    v2f aext;
    aext.x = (hi == 0) ? 1.0f : 0.0f;  // K=64 -> constant 1 (linear term)
    aext.y = 0.0f;                     // K=65/67 -> zero pad
    a[16] = aext;
  }

  __syncthreads();

  // ---- Z = X * Bmat via v_wmma_f32_16x16x4_f32, fused with out = sum_i x_i*z_i
  float part[8];
  #pragma unroll
  for (int v = 0; v < 8; ++v) part[v] = 0.f;

  #pragma unroll
  for (int t = 0; t < 4; ++t) {        // i-tiles: i = 16t + lo
    v8f acc = {};
    #pragma unroll
    for (int kk = 0; kk < kKChunks; ++kk) {
      v2f b = Bp[(2 * kk + hi) * kBpStride + (16 * t + lo)];
      acc = __builtin_amdgcn_wmma_f32_16x16x4_f32(
          /*neg_a=*/false, a[kk], /*neg_b=*/false, b,
          /*c_mod=*/(short)0, acc, /*reuse_a=*/false, /*reuse_b=*/false);
    }
    // D layout: lanes 0-15 -> M=v, lanes 16-31 -> M=v+8; N = lo; i = 16t+lo
    #pragma unroll
    for (int v = 0; v < 8; ++v) {
      const int mr = v + 8 * hi;
      part[v] += Xs[wave][mr * kXsStride + 16 * t + lo] * acc[v];
    }
  }

  // ---- butterfly reduce over the 16 lanes of each half-wave ----
  #pragma unroll
  for (int v = 0; v < 8; ++v) {
    float s = part[v];
    s += __shfl_xor(s, 1, 32);
    s += __shfl_xor(s, 2, 32);
    s += __shfl_xor(s, 4, 32);
    s += __shfl_xor(s, 8, 32);
    part[v] = s;
  }

  const float b0 = bias[h];
  #pragma unroll
  for (int v = 0; v < 8; ++v) {
    if (lo == v) {                       // lanes (v,hi=0) and (v,hi=1) write
      const int row = n_base + v + 8 * hi;
      out[(size_t)row * kH + h] = part[v] + b0;
    }
  }
}

extern "C" void kernel_launch(void* const* d_in, const int* in_sizes, int n_in,
                              void* d_out, int out_size, void* d_ws, size_t ws_size,
                              hipStream_t stream) {
  const float* obs  = (const float*)d_in[0];  // [N, 64, 64] f32
  const float* W    = (const float*)d_in[1];  // [64, 2145]  f32
  const float* bias = (const float*)d_in[2];  // [64]        f32
  float* out = (float*)d_out;                 // [N, 64]     f32

  const int num_paths = in_sizes[0] / (kH * kD);            // 1024
  dim3 grid(num_paths / (kWaves * kRowsPerWave), kH);       // (8, 64)
  qvf_wmma_kernel<<<grid, 256, 0, stream>>>(obs, W, bias, out);
}